// hermite_poly_act_4750233829891
// MI455X (gfx1250) — compile-verified
//
#include <hip/hip_runtime.h>

// ---------------------------------------------------------------------------
// out[b,c] = sum_k params[c,k] * H_k(x[b,c])   (normalized probabilists' Hermite)
// Rewritten as out[b,c] = sum_j q[c,j] * x^j with q = params @ M^T  (M = Hermite->
// monomial change of basis). Kernel 1 computes q with V_WMMA_F32_16X16X4_F32
// (a genuine tiny GEMM, K padded 11->16). Kernel 2 is the bandwidth-bound
// streaming Horner evaluation (10 FMAs/element), which is the roofline limiter.
// ---------------------------------------------------------------------------

typedef float v2f __attribute__((ext_vector_type(2)));
typedef float v4f __attribute__((ext_vector_type(4)));
typedef float v8f __attribute__((ext_vector_type(8)));

// Monomial coefficients of unnormalized probabilists' Hermite polys H_0..H_10.
__device__ __constant__ int c_hc[11][11] = {
    {   1,    0,     0,     0,     0,   0,   0,   0,  0, 0, 0},  // H0
    {   0,    1,     0,     0,     0,   0,   0,   0,  0, 0, 0},  // H1
    {  -1,    0,     1,     0,     0,   0,   0,   0,  0, 0, 0},  // H2
    {   0,   -3,     0,     1,     0,   0,   0,   0,  0, 0, 0},  // H3
    {   3,    0,    -6,     0,     1,   0,   0,   0,  0, 0, 0},  // H4
    {   0,   15,     0,   -10,     0,   1,   0,   0,  0, 0, 0},  // H5
    { -15,    0,    45,     0,   -15,   0,   1,   0,  0, 0, 0},  // H6
    {   0, -105,     0,   105,     0, -21,   0,   1,  0, 0, 0},  // H7
    { 105,    0,  -420,     0,   210,   0, -28,   0,  1, 0, 0},  // H8
    {   0,  945,     0, -1260,     0, 378,   0, -36,  0, 1, 0},  // H9
    {-945,    0,  4725,     0, -3150,   0, 630,   0,-45, 0, 1},  // H10
};

// 1/sqrt(k!) normalizers
__device__ __constant__ float c_rsqfact[11] = {
    1.0f, 1.0f, 0.70710678f, 0.40824829f, 0.20412415f, 0.091287093f,
    0.037267800f, 0.014085904f, 0.0049801190f, 0.0016600400f, 0.00052495195f};

__device__ __forceinline__ float mt_val(int k, int j) {
    // M^T entry: coefficient of x^j in normalized H_k; zero-padded to 16x16.
    if (k > 10 || j > 10) return 0.0f;
    return (float)c_hc[k][j] * c_rsqfact[k];
}

// ---------------------------------------------------------------------------
// Kernel 1: q[16 x 16 tile] = params_tile[16 x 16(K pad)] x M^T[16 x 16]
// via 4 chained V_WMMA_F32_16X16X4_F32 (K=4 each). One wave32 per 16 channels.
// A layout (16x4 f32): lanes 0-15 hold rows 0-15 with {K0,K1}; lanes 16-31 {K2,K3}.
// C/D layout: VGPR i holds row i (lanes 0-15, N=lane) / row i+8 (lanes 16-31).
// ---------------------------------------------------------------------------
__global__ void __launch_bounds__(32)
hermite_q_wmma(const float* __restrict__ params, float* __restrict__ qws) {
    const int lane  = threadIdx.x;       // 0..31
    const int half  = lane >> 4;         // 0 or 1
    const int l     = lane & 15;
    const int cbase = blockIdx.x * 16;

    v8f acc = {};
#pragma unroll
    for (int s = 0; s < 4; ++s) {
        const int k0 = 4 * s + 2 * half;
        const int k1 = k0 + 1;
        // A fragment: params row (channel cbase+l), K columns k0,k1 (zero-pad K>=11)
        const size_t rowOff = (size_t)(cbase + l) * 11;
        v2f a;
        a.x = params[rowOff + (k0 < 11 ? k0 : 10)] * (k0 < 11 ? 1.0f : 0.0f);
        a.y = params[rowOff + (k1 < 11 ? k1 : 10)] * (k1 < 11 ? 1.0f : 0.0f);
        // B fragment: M^T rows k0,k1 at column n=l
        v2f b;
        b.x = mt_val(k0, l);
        b.y = mt_val(k1, l);
        acc = __builtin_amdgcn_wmma_f32_16x16x4_f32(
            /*neg_a=*/false, a, /*neg_b=*/false, b,
            /*c_mod=*/(short)0, acc, /*reuse_a=*/false, /*reuse_b=*/false);
    }
    // Store D: row = i + 8*half, col = l. q row stride padded to 16 floats.
#pragma unroll
    for (int i = 0; i < 8; ++i) {
        const int m = i + 8 * half;
        qws[(size_t)(cbase + m) * 16 + l] = acc[i];
    }
}

// ---------------------------------------------------------------------------
// Kernel 2: streaming Horner. Thread owns 4 consecutive channels (float4 lane
// accesses -> 512B/wave coalesced) and ROWS_PER_BLOCK rows of b. Nontemporal:
// x and out are single-touch and together slightly exceed the 192MB L2.
// ---------------------------------------------------------------------------
#define H_TPB  256
#define H_CPT  4
#define H_ROWS 32

__device__ __forceinline__ void horner_stream(const float* __restrict__ x,
                                              float* __restrict__ out,
                                              const float qc[H_CPT][11],
                                              int cvec, int Cv, int b0) {
    const v4f* __restrict__ xv = (const v4f*)x;
    v4f* __restrict__ ov = (v4f*)out;
#pragma unroll 2
    for (int r = 0; r < H_ROWS; ++r) {
        const size_t idx = (size_t)(b0 + r) * Cv + cvec;
        v4f xq = __builtin_nontemporal_load(&xv[idx]);
        float xs[H_CPT] = {xq.x, xq.y, xq.z, xq.w};
        float os[H_CPT];
#pragma unroll
        for (int ch = 0; ch < H_CPT; ++ch) {
            float acc = qc[ch][10];
#pragma unroll
            for (int j = 9; j >= 0; --j) acc = fmaf(acc, xs[ch], qc[ch][j]);
            os[ch] = acc;
        }
        v4f o;
        o.x = os[0]; o.y = os[1]; o.z = os[2]; o.w = os[3];
        __builtin_nontemporal_store(o, &ov[idx]);
    }
}

__global__ void __launch_bounds__(H_TPB)
hermite_horner(const float* __restrict__ x, const float* __restrict__ qws,
               float* __restrict__ out, int C) {
    const int cvec = blockIdx.x * blockDim.x + threadIdx.x;  // float4 index in C
    const int c4   = cvec * H_CPT;
    const int Cv   = C >> 2;
    const int b0   = blockIdx.y * H_ROWS;

    float qc[H_CPT][11];
#pragma unroll
    for (int ch = 0; ch < H_CPT; ++ch) {
        const v4f* qr = (const v4f*)(qws + (size_t)(c4 + ch) * 16);
        v4f a = qr[0], b = qr[1], c = qr[2];
        qc[ch][0] = a.x; qc[ch][1] = a.y; qc[ch][2]  = a.z; qc[ch][3] = a.w;
        qc[ch][4] = b.x; qc[ch][5] = b.y; qc[ch][6]  = b.z; qc[ch][7] = b.w;
        qc[ch][8] = c.x; qc[ch][9] = c.y; qc[ch][10] = c.z;
    }
    horner_stream(x, out, qc, cvec, Cv, b0);
}

// Fallback if workspace is too small: derive q[c,:] in-thread (amortized over
// H_ROWS rows, negligible), then identical streaming body.
__global__ void __launch_bounds__(H_TPB)
hermite_horner_inline(const float* __restrict__ x, const float* __restrict__ params,
                      float* __restrict__ out, int C) {
    const int cvec = blockIdx.x * blockDim.x + threadIdx.x;
    const int c4   = cvec * H_CPT;
    const int Cv   = C >> 2;
    const int b0   = blockIdx.y * H_ROWS;

    float qc[H_CPT][11];
#pragma unroll
    for (int ch = 0; ch < H_CPT; ++ch) {
        float p[11];
#pragma unroll
        for (int k = 0; k < 11; ++k) p[k] = params[(size_t)(c4 + ch) * 11 + k];
        for (int j = 0; j < 11; ++j) {
            float s = 0.0f;
            for (int k = j; k < 11; ++k) s = fmaf(p[k], mt_val(k, j), s);
            qc[ch][j] = s;
        }
    }
    horner_stream(x, out, qc, cvec, Cv, b0);
}

extern "C" void kernel_launch(void* const* d_in, const int* in_sizes, int n_in,
                              void* d_out, int out_size, void* d_ws, size_t ws_size,
                              hipStream_t stream) {
    const float* x      = (const float*)d_in[0];
    const float* params = (const float*)d_in[1];
    float*       out    = (float*)d_out;

    const int CK = in_sizes[1];   // C * 11
    const int C  = CK / 11;       // 4096
    const int BC = in_sizes[0];   // B * C
    const int B  = BC / C;        // 8192

    const size_t qBytes = (size_t)C * 16 * sizeof(float);
    float* qws = (float*)d_ws;
    const bool useWs = (qws != nullptr) && (ws_size >= qBytes);

    const int gx = C / (H_TPB * H_CPT);  // 4
    const int gy = B / H_ROWS;           // 256
    dim3 grid(gx, gy), block(H_TPB);

    if (useWs) {
        hermite_q_wmma<<<C / 16, 32, 0, stream>>>(params, qws);
        hermite_horner<<<grid, block, 0, stream>>>(x, qws, out, C);
    } else {
        hermite_horner_inline<<<grid, block, 0, stream>>>(x, params, out, C);
    }
}